// Attention_28321014349957
// MI455X (gfx1250) — compile-verified
//
#include <hip/hip_runtime.h>

typedef __attribute__((ext_vector_type(2))) float v2f;
typedef __attribute__((ext_vector_type(8))) float v8f;

#define IMG_W   256
#define IMG_HW  (256 * 256)
#define NBATCH  8
#define CDIM    64                    // channels
#define C3      192                   // 3*CDIM (qkv channels)

// ----- dynamic LDS layout (147 KB total) -----------------------------------
// sx   : x halo, [64][112]  (100 valid halo pixels + 12 zero-pad columns)
//        row stride 112: 2*112 % 64 == 32  -> WMMA-B half-waves bank-disjoint
// sqh  : qkv halo, [192][116]
//        row stride 116: 8*116 % 64 == 32  -> WMMA-D half-wave stores disjoint
// sqk  : dw'd q (rows 0..63) / k (rows 64..127), [128][64]
// sattn: attention result, [64][72] (overlays sx once sx is dead)
//        row stride 72: 2*72 % 64 == 16    -> WMMA-B half-waves bank-disjoint
#define SX_STRIDE   112
#define SX_BYTES    (64 * 112 * 4)                    // 28672
#define SQH_STRIDE  116
#define SQH_BYTES   (192 * 116 * 4)                   // 89088
#define SQK_BYTES   (128 * 64 * 4)                    // 32768
#define SATT_STRIDE 72
#define SMEM_BYTES  (SX_BYTES + SQH_BYTES + SQK_BYTES)  // 150528 (147 KB)

// ---------------------------------------------------------------------------
// One 256-thread workgroup per (batch, 8x8 patch). Entire op fused; the only
// HBM traffic is reading x (L2-resident, 134 MB < 192 MB L2) and writing out.
// ---------------------------------------------------------------------------
__global__ __launch_bounds__(256)
void fused_attention(const float* __restrict__ X,     // [8,64,256,256]
                     const float* __restrict__ QW,    // [192,64] qkv 1x1 w
                     const float* __restrict__ W3,    // [192,1,3,3] dw w
                     const float* __restrict__ PW,    // [64,64] proj 1x1 w
                     const float* __restrict__ temp,  // [64]
                     float* __restrict__ Out) {       // [8,64,256,256]
  extern __shared__ char smem[];
  float* sx    = (float*)smem;
  float* sattn = (float*)smem;                        // overlays sx (dead)
  float* sqh   = (float*)(smem + SX_BYTES);
  float* sqk   = (float*)(smem + SX_BYTES + SQH_BYTES);

  const int tid   = threadIdx.x;
  const int patch = blockIdx.x;                       // 0..1023
  const int b     = blockIdx.y;
  const int ph = patch >> 5, pwi = patch & 31;
  const int y0 = ph * 8,     x0  = pwi * 8;

  const int lane = tid & 31, wave = tid >> 5;
  const int half = lane >> 4, sub = lane & 15;        // WMMA lane roles

  // ---- stage 0: x halo (10x10, all 64 ch) -> LDS, zero-padded -------------
  // Out-of-image zeros propagate as exact zeros through the 1x1 conv, which
  // reproduces the reference's SAME zero padding for the depthwise conv.
  const float* Xb = X + (size_t)b * CDIM * IMG_HW;
  for (int i = 0; i < 28; ++i) {
    const int idx = i * 256 + tid;                    // 0..7167 == c*112 + j
    const int c = idx / SX_STRIDE;
    const int j = idx % SX_STRIDE;
    float v = 0.f;
    if (j < 100) {
      const int y = y0 - 1 + j / 10;
      const int x = x0 - 1 + j % 10;
      if (y >= 0 && y < 256 && x >= 0 && x < 256)
        v = Xb[(size_t)c * IMG_HW + (size_t)y * IMG_W + x];
    }
    sx[idx] = v;
  }
  __syncthreads();

  // ---- stage 1: qkv halo GEMM  QW(192x64) x sx(64x112) -> sqh -------------
  // 12 M-tiles x 7 N-tiles = 84 WMMA tiles over 8 waves; A from global
  // (49 KB, L2-hot), B ds_load'ed from LDS, D ds_store'd to LDS.
  for (int tile = wave; tile < 84; tile += 8) {
    const int m0 = (tile / 7) * 16;
    const int n0 = (tile % 7) * 16;
    v8f acc = {};
#pragma unroll
    for (int k = 0; k < CDIM; k += 4) {
      const int kb = k + 2 * half;
      v2f a = *(const v2f*)(QW + (size_t)(m0 + sub) * CDIM + kb);
      v2f bf;
      bf.x = sx[kb       * SX_STRIDE + n0 + sub];
      bf.y = sx[(kb + 1) * SX_STRIDE + n0 + sub];
      acc = __builtin_amdgcn_wmma_f32_16x16x4_f32(
          false, a, false, bf, (short)0, acc, false, false);
    }
#pragma unroll
    for (int r = 0; r < 8; ++r)
      sqh[(m0 + r + 8 * half) * SQH_STRIDE + n0 + sub] = acc[r];
  }
  __syncthreads();

  // ---- stage 2: depthwise 3x3 from LDS halo (no bounds checks needed) -----
  // q,k -> sqk; v -> registers of the thread that consumes it in stage 3
  // ((128+c)*64+pix == c*64+pix (mod 256) => same thread, iteration i-32).
  float vreg[16];
  for (int i = 0; i < 48; ++i) {
    const int idx = i * 256 + tid;                    // 0..12287
    const int c3  = idx >> 6;                         // qkv channel 0..191
    const int pix = idx & 63;
    const int ty = pix >> 3, tx = pix & 7;
    const float* w = W3 + c3 * 9;
    float s = 0.f;
#pragma unroll
    for (int ky = 0; ky < 3; ++ky)
#pragma unroll
      for (int kx = 0; kx < 3; ++kx)
        s = fmaf(sqh[c3 * SQH_STRIDE + (ty + ky) * 10 + tx + kx],
                 w[ky * 3 + kx], s);
    if (c3 < 2 * CDIM) sqk[c3 * 64 + pix] = s;
    else               vreg[i - 32] = s;
  }
  __syncthreads();

  // ---- stage 3: 8x8 circular conv (== irfft2(rfft2 q * rfft2 k)) ----------
  // then * temperature[c] * v  -> sattn (K=channel x N=pixel for stage 4)
  for (int i = 0; i < 16; ++i) {
    const int idx = i * 256 + tid;                    // 0..4095
    const int c   = idx >> 6;
    const int pix = idx & 63;
    const int ty = pix >> 3, tx = pix & 7;
    float s = 0.f;
#pragma unroll
    for (int n1 = 0; n1 < 8; ++n1)
#pragma unroll
      for (int n2 = 0; n2 < 8; ++n2)
        s = fmaf(sqk[c * 64 + n1 * 8 + n2],
                 sqk[(CDIM + c) * 64 + ((ty - n1) & 7) * 8 + ((tx - n2) & 7)],
                 s);
    sattn[c * SATT_STRIDE + pix] = s * temp[c] * vreg[i];
  }
  __syncthreads();

  // ---- stage 4: proj GEMM  PW(64x64) x sattn(64x64) -> Out (WMMA) ---------
#pragma unroll
  for (int t = 0; t < 2; ++t) {
    const int tile = wave * 2 + t;                    // 0..15
    const int m0 = (tile >> 2) * 16;
    const int n0 = (tile & 3)  * 16;
    v8f acc = {};
#pragma unroll
    for (int k = 0; k < CDIM; k += 4) {
      const int kb = k + 2 * half;
      v2f a = *(const v2f*)(PW + (size_t)(m0 + sub) * CDIM + kb);  // L2-hot
      v2f bf;
      bf.x = sattn[kb       * SATT_STRIDE + n0 + sub];
      bf.y = sattn[(kb + 1) * SATT_STRIDE + n0 + sub];
      acc = __builtin_amdgcn_wmma_f32_16x16x4_f32(
          false, a, false, bf, (short)0, acc, false, false);
    }
#pragma unroll
    for (int r = 0; r < 8; ++r) {
      const int m = m0 + r + 8 * half;
      const int n = n0 + sub;
      Out[((size_t)b * CDIM + m) * IMG_HW +
          (size_t)(y0 + (n >> 3)) * IMG_W + (x0 + (n & 7))] = acc[r];
    }
  }
}

// ---------------------------------------------------------------------------
// Single fully-fused launch. No workspace: qkv / dw-qkv / attn intermediates
// live entirely in LDS; HBM sees only x (reads, L2-resident) and the output.
// ---------------------------------------------------------------------------
extern "C" void kernel_launch(void* const* d_in, const int* in_sizes, int n_in,
                              void* d_out, int out_size, void* d_ws, size_t ws_size,
                              hipStream_t stream) {
  const float* x      = (const float*)d_in[0];  // [8,64,256,256]
  const float* qkv_w  = (const float*)d_in[1];  // [192,64,1,1]
  const float* dw_w   = (const float*)d_in[2];  // [192,1,3,3]
  const float* proj_w = (const float*)d_in[3];  // [64,64,1,1]
  const float* temp   = (const float*)d_in[4];  // [64,1,1]
  float* out = (float*)d_out;

  fused_attention<<<dim3(1024, NBATCH), 256, SMEM_BYTES, stream>>>(
      x, qkv_w, dw_w, proj_w, temp, out);
}